// Conv_34187939676169
// MI455X (gfx1250) — compile-verified
//
#include <hip/hip_runtime.h>
#include <hip/hip_bf16.h>

// The whole depth-8 conv stack on 3x3 single-channel images collapses to a
// single affine map out = M*x + c with M 9x9, c 9-vector. Prep kernel composes
// M,c from the 8 layers; main kernel streams 4M samples through
// V_WMMA_F32_16X16X4_F32 (16 samples per tile, K=9 padded to 12 -> 3 WMMAs).
// Memory-bound: 288 MB total traffic -> ~12.4 us floor at 23.3 TB/s.

typedef __attribute__((ext_vector_type(2))) float v2f;
typedef __attribute__((ext_vector_type(8))) float v8f;

// ---- workspace layout (floats) ----
// [0   .. 191]  Bpack[t][r][lane]  (t=0..2 K-chunk, r=0..1 VGPR, lane=0..31)
//               value = M[n][k], n=lane&15, k=4t+2*(lane>>4)+r  (0 if OOR)
// [192 .. 207]  cvec[16]  (c[n] for n<9 else 0) -> accumulator init
// [208 .. 288]  M row-major (81)   (scalar remainder path)
// [289 .. 297]  c (9)
#define WS_BPACK 0
#define WS_CVEC  192
#define WS_M     208
#define WS_C     289

__global__ void conv_stack_prep_kernel(const float* __restrict__ W,
                                       const float* __restrict__ bias,
                                       float* __restrict__ ws) {
  if (blockIdx.x != 0 || threadIdx.x != 0) return;
  float M[9][9], c[9];
  for (int p = 0; p < 9; ++p) {
    c[p] = 0.0f;
    for (int q = 0; q < 9; ++q) M[p][q] = (p == q) ? 1.0f : 0.0f;
  }
  for (int d = 0; d < 8; ++d) {
    const float* w = W + d * 9;          // weights[d,0,0,ky,kx]
    float T[9][9];
    for (int p = 0; p < 9; ++p) {
      int pi = p / 3, pj = p % 3;
      for (int q = 0; q < 9; ++q) {
        int qi = q / 3, qj = q % 3;
        int ky = qi - pi + 1, kx = qj - pj + 1;
        T[p][q] = (ky >= 0 && ky < 3 && kx >= 0 && kx < 3) ? w[ky * 3 + kx] : 0.0f;
      }
    }
    float nM[9][9], nc[9];
    for (int p = 0; p < 9; ++p) {
      float s = 0.0f;
      for (int q = 0; q < 9; ++q) s += T[p][q] * c[q];
      nc[p] = s + bias[d];
      for (int r = 0; r < 9; ++r) {
        float m = 0.0f;
        for (int q = 0; q < 9; ++q) m += T[p][q] * M[q][r];
        nM[p][r] = m;
      }
    }
    for (int p = 0; p < 9; ++p) {
      c[p] = nc[p];
      for (int q = 0; q < 9; ++q) M[p][q] = nM[p][q];
    }
  }
  // B-matrix lane image: B[k][n] = M[n][k]; lanes 0-15 hold N=0-15 with
  // VGPR0=K(4t+0),VGPR1=K(4t+1); lanes 16-31 hold K(4t+2),K(4t+3).
  for (int t = 0; t < 3; ++t)
    for (int r = 0; r < 2; ++r)
      for (int lane = 0; lane < 32; ++lane) {
        int n = lane & 15, h = lane >> 4;
        int k = 4 * t + 2 * h + r;
        ws[WS_BPACK + (t * 2 + r) * 32 + lane] =
            (n < 9 && k < 9) ? M[n][k] : 0.0f;
      }
  for (int n = 0; n < 16; ++n) ws[WS_CVEC + n] = (n < 9) ? c[n] : 0.0f;
  for (int p = 0; p < 9; ++p)
    for (int q = 0; q < 9; ++q) ws[WS_M + p * 9 + q] = M[p][q];
  for (int p = 0; p < 9; ++p) ws[WS_C + p] = c[p];
}

__global__ void __launch_bounds__(256)
conv_stack_affine_wmma_kernel(const float* __restrict__ x,
                              float* __restrict__ out,
                              const float* __restrict__ ws,
                              int ntiles, int rem, int nsamp) {
  const int lane = threadIdx.x & 31;
  const int s = lane & 15;      // sample-in-tile (A rows / D cols)
  const int h = lane >> 4;      // half-wave -> K pair select
  const int wavesPerBlock = blockDim.x >> 5;
  const int wave = blockIdx.x * wavesPerBlock + (threadIdx.x >> 5);
  const int nwaves = gridDim.x * wavesPerBlock;

  // Wave-invariant B operands (M^T lane images) + bias accumulator init.
  v2f b0, b1, b2;
  b0.x = ws[WS_BPACK + 0 * 32 + lane];  b0.y = ws[WS_BPACK + 1 * 32 + lane];
  b1.x = ws[WS_BPACK + 2 * 32 + lane];  b1.y = ws[WS_BPACK + 3 * 32 + lane];
  b2.x = ws[WS_BPACK + 4 * 32 + lane];  b2.y = ws[WS_BPACK + 5 * 32 + lane];
  const float cv = ws[WS_CVEC + s];

  for (int tile = wave; tile < ntiles; tile += nwaves) {
    const float* xb = x + (size_t)tile * 144;   // 16 samples * 9 floats
    if (tile + nwaves < ntiles)
      __builtin_prefetch(x + (size_t)(tile + nwaves) * 144 + s * 9, 0, 0);

    // A-matrix 16x4 per chunk: lane s holds A[s][2h], A[s][2h+1].
    const int base = s * 9 + 2 * h;
    v2f a0, a1, a2;
    a0.x = xb[base + 0];  a0.y = xb[base + 1];       // k = 2h, 2h+1
    a1.x = xb[base + 4];  a1.y = xb[base + 5];       // k = 4+2h, 5+2h
    const float x8 = xb[s * 9 + 8];                  // k = 8 (only valid tap)
    a2.x = h ? 0.0f : x8;                            // k = 8+2h (pad k>=9)
    a2.y = 0.0f;

    v8f acc = {cv, cv, cv, cv, cv, cv, cv, cv};      // bias pre-loaded in C
    acc = __builtin_amdgcn_wmma_f32_16x16x4_f32(false, a0, false, b0,
                                                (short)0, acc, false, false);
    acc = __builtin_amdgcn_wmma_f32_16x16x4_f32(false, a1, false, b1,
                                                (short)0, acc, false, false);
    acc = __builtin_amdgcn_wmma_f32_16x16x4_f32(false, a2, false, b2,
                                                (short)0, acc, false, false);

    // D layout: VGPR r -> sample (r + 8h), feature s. Store 9 valid columns.
    float* ob = out + (size_t)tile * 144;
    if (s < 9) {
#pragma unroll
      for (int r = 0; r < 8; ++r) ob[(r + 8 * h) * 9 + s] = acc[r];
    }
  }

  // Remainder samples (N % 16): plain scalar affine map (N=4M -> rem==0).
  const int gtid = blockIdx.x * blockDim.x + threadIdx.x;
  if (gtid < rem) {
    const int idx = ntiles * 16 + gtid;
    if (idx < nsamp) {
      const float* xi = x + (size_t)idx * 9;
      float xv[9];
#pragma unroll
      for (int q = 0; q < 9; ++q) xv[q] = xi[q];
      float* oi = out + (size_t)idx * 9;
#pragma unroll
      for (int p = 0; p < 9; ++p) {
        float acc = ws[WS_C + p];
#pragma unroll
        for (int q = 0; q < 9; ++q) acc = fmaf(ws[WS_M + p * 9 + q], xv[q], acc);
        oi[p] = acc;
      }
    }
  }
}

extern "C" void kernel_launch(void* const* d_in, const int* in_sizes, int n_in,
                              void* d_out, int out_size, void* d_ws, size_t ws_size,
                              hipStream_t stream) {
  const float* x = (const float*)d_in[0];     // [N, 9] f32
  const float* W = (const float*)d_in[1];     // [8,1,1,3,3] f32
  const float* b = (const float*)d_in[2];     // [8] f32
  float* out = (float*)d_out;
  float* ws = (float*)d_ws;                   // needs ~1.2 KB

  const int nsamp = in_sizes[0] / 9;
  const int ntiles = nsamp / 16;
  const int rem = nsamp % 16;

  conv_stack_prep_kernel<<<1, 64, 0, stream>>>(W, b, ws);

  int blocks = (ntiles + 7) / 8;              // 8 waves per 256-thread block
  if (blocks > 2048) blocks = 2048;
  if (blocks < 1) blocks = 1;
  conv_stack_affine_wmma_kernel<<<blocks, 256, 0, stream>>>(x, out, ws,
                                                            ntiles, rem, nsamp);
}